// GEP_74174085202409
// MI455X (gfx1250) — compile-verified
//
#include <hip/hip_runtime.h>

// ---------------------------------------------------------------------------
// GEP (gradient embedding projection) for MI455X / gfx1250, f32 WMMA path.
// Round 4: 32-bit element offsets off uniform kernel-arg bases everywhere
// (global_load saddr form, no flat_load / DScnt coupling, no 64-bit per-lane
// address adds), use-then-prefetch loop with unroll-2 ping-pong buffers.
// ---------------------------------------------------------------------------

typedef __attribute__((ext_vector_type(2))) float v2f;
typedef __attribute__((ext_vector_type(8))) float v8f;

#define BATCH   64
#define PTOT    500000
#define KTOT    398
#define ESTRIDE 416              // padded row stride for E (13 x 128B lines)
#define CLIP0   1.0f
#define CLIP1   1.0f

// workspace layout in floats
#define WS_E        0
#define WS_RESNORM  (BATCH * ESTRIDE)          // 26624
#define WS_RESSCALE (WS_RESNORM + BATCH)       // 26688
#define WS_FLOATS   (WS_RESSCALE + BATCH)      // 26752 floats (~107 KB)

__device__ __forceinline__ v8f v8f_zero() {
    v8f z = {0.f, 0.f, 0.f, 0.f, 0.f, 0.f, 0.f, 0.f};
    return z;
}

// D = A(16x4) * B(4x16) + C, f32 WMMA.
__device__ __forceinline__ v8f wmma_f32(v2f a, v2f b, v8f c) {
    return __builtin_amdgcn_wmma_f32_16x16x4_f32(
        false, a, false, b, (short)0, c, false, false);
}

// ---------------------------------------------------------------------------
// Phase 0: zero E buffer + residual norm accumulators
// ---------------------------------------------------------------------------
__global__ void k_init(float* __restrict__ ws) {
    int i = blockIdx.x * blockDim.x + threadIdx.x;
    if (i < WS_FLOATS) ws[i] = 0.f;
}

// ---------------------------------------------------------------------------
// Phase 1: E[b,k] += sum_p G[b, goff+p] * Bg[p,k]   (split-K over P, atomics)
// Block = 256 threads (8 waves). wave: mt = w&3 (M tile), half = w>>2,
// owns N tiles nt = half*CNT + j, j < CNT (compile-time).
// Out-of-range tiles/columns are address-clamped in the hot loop and
// discarded in the epilogue (garbage D columns never escape).
// ---------------------------------------------------------------------------
template <int KG>
__global__ __launch_bounds__(256) void k_gemm1(
    const float* __restrict__ G, const float* __restrict__ Bg,
    float* __restrict__ wsE,
    int Pg, int goff, int koff, int chunk)
{
    constexpr int NT  = (KG + 15) >> 4;        // N tiles
    constexpr int CNT = (NT + 1) >> 1;         // tiles per wave (4 or 5)

    const int tid  = threadIdx.x;
    const int w    = tid >> 5;
    const int lane = tid & 31;
    const int mt   = w & 3;
    const int half = w >> 2;
    const int lo   = lane & 15;
    const int hi   = lane >> 4;
    const int kk   = hi * 2;                   // K offset within the 4-step

    const int p0 = blockIdx.x * chunk;
    const int p1 = min(p0 + chunk, Pg);        // Pg % 4 == 0 for all groups
    const int nsteps = (p1 - p0) >> 2;

    const int m = mt * 16 + lo;                // A-row (batch index)

    // per-tile clamped column + write predicate (loop-invariant)
    int  colc[CNT];
    bool wr[CNT];
#pragma unroll
    for (int j = 0; j < CNT; ++j) {
        int nt  = half * CNT + j;
        int ntc = min(nt, NT - 1);
        int col = ntc * 16 + lo;
        colc[j] = min(col, KG - 1);
        wr[j]   = (nt < NT) && (col < KG);
    }

    // 32-bit element offsets off the uniform base pointers (GVS addressing)
    unsigned aoff = (unsigned)m * PTOT + (unsigned)(goff + p0 + kk);
    unsigned boff[CNT];
#pragma unroll
    for (int j = 0; j < CNT; ++j)
        boff[j] = (unsigned)(p0 + kk) * KG + (unsigned)colc[j];

    v8f acc[CNT];
#pragma unroll
    for (int j = 0; j < CNT; ++j) acc[j] = v8f_zero();

    if (nsteps > 0) {
        v2f a = *(const v2f*)(G + aoff);
        v2f b[CNT];
#pragma unroll
        for (int j = 0; j < CNT; ++j) {
            b[j].x = Bg[boff[j]];
            b[j].y = Bg[boff[j] + KG];         // constant IOFFSET
        }

#pragma unroll 2
        for (int s = 0; s < nsteps; ++s) {
            // consume current fragments
#pragma unroll
            for (int j = 0; j < CNT; ++j)
                acc[j] = wmma_f32(a, b[j], acc[j]);

            // advance + prefetch next (independent of the WMMAs above;
            // the scheduler can hoist these across the matrix ops)
            aoff += 4;
#pragma unroll
            for (int j = 0; j < CNT; ++j) boff[j] += 4u * KG;
            if (s + 1 < nsteps) {              // scalar-uniform
                a = *(const v2f*)(G + aoff);
#pragma unroll
                for (int j = 0; j < CNT; ++j) {
                    b[j].x = Bg[boff[j]];
                    b[j].y = Bg[boff[j] + KG];
                }
            }
        }
    }

#pragma unroll
    for (int j = 0; j < CNT; ++j) {
        if (wr[j]) {
#pragma unroll
            for (int v = 0; v < 8; ++v) {
                int row = mt * 16 + v + hi * 8;
                atomicAdd(&wsE[row * ESTRIDE + koff + colc[j]], acc[j][v]);
            }
        }
    }
}

// ---------------------------------------------------------------------------
// Phase 2: clip rows of E, batch-average -> out[0..397]
// ---------------------------------------------------------------------------
__global__ __launch_bounds__(256) void k_embfinish(
    const float* __restrict__ wsE, float* __restrict__ out)
{
    __shared__ float sEmb[BATCH];
    const int t = threadIdx.x;
    if (t < BATCH) {
        float n2 = 0.f;
        const float* row = wsE + t * ESTRIDE;
        for (int k = 0; k < KTOT; ++k) { float e = row[k]; n2 += e * e; }
        sEmb[t] = fminf(CLIP0 / sqrtf(n2), 1.0f);   // norm==0 -> inf -> 1
    }
    __syncthreads();
    for (int k = t; k < KTOT; k += 256) {
        float sum = 0.f;
        for (int b = 0; b < BATCH; ++b) sum += sEmb[b] * wsE[b * ESTRIDE + k];
        out[k] = sum * (1.0f / BATCH);
    }
}

// ---------------------------------------------------------------------------
// Shared back-projection accumulator: approx tile = E_slice @ Bg^T
// Block tile: 64 batch x 64 P-columns; wave: mt = w&3, half = w>>2,
// owns nt = half and half+2 (NT == 4 always).
// K-loop compile-time bounded, fully unrolled, constant-k IOFFSET folding.
// A comes from the zero-padded E buffer so out-of-range K terms vanish;
// B rows clamped to Pg-1 (invalid D columns discarded by callers).
// ---------------------------------------------------------------------------
template <int KG>
__device__ __forceinline__ void backproj_acc(
    const float* __restrict__ wsE, const float* __restrict__ Bg,
    int Pg, int koff, int pbase,
    int mt, int half, int lo, int hi, v8f acc[2])
{
    constexpr int KG4 = KG & ~3;

    const int m  = mt * 16 + lo;
    const int kk = hi * 2;

    const unsigned eoff = (unsigned)m * ESTRIDE + (unsigned)(koff + kk);
    const int pA = pbase + half * 16 + lo;
    const int pB = pA + 32;
    const unsigned b0off = (unsigned)min(pA, Pg - 1) * KG + (unsigned)kk;
    const unsigned b1off = (unsigned)min(pB, Pg - 1) * KG + (unsigned)kk;

#pragma unroll
    for (int k = 0; k < KG4; k += 4) {
        v2f a  = *(const v2f*)(wsE + eoff + k);    // constant IOFFSET
        v2f b0 = *(const v2f*)(Bg + b0off + k);
        v2f b1 = *(const v2f*)(Bg + b1off + k);
        acc[0] = wmma_f32(a, b0, acc[0]);
        acc[1] = wmma_f32(a, b1, acc[1]);
    }

    if constexpr (KG4 < KG) {                  // tail step (KG=138 only)
        constexpr int REM = KG - KG4;          // == 2
        v2f a = *(const v2f*)(wsE + eoff + KG4);   // padded E: zeros beyond KG
        v2f b0, b1;
        if (kk < REM) {                        // kk==0 half: cols valid
            b0 = *(const v2f*)(Bg + b0off + KG4);
            b1 = *(const v2f*)(Bg + b1off + KG4);
        } else {                               // kk==2 half: beyond KG -> A==0
            b0.x = 0.f; b0.y = 0.f;
            b1.x = 0.f; b1.y = 0.f;
        }
        acc[0] = wmma_f32(a, b0, acc[0]);
        acc[1] = wmma_f32(a, b1, acc[1]);
    }
}

// ---------------------------------------------------------------------------
// Phase 3: residual norms^2 per batch row (pass 1 of back-projection)
// ---------------------------------------------------------------------------
template <int KG>
__global__ __launch_bounds__(256) void k_resnorm(
    const float* __restrict__ G, const float* __restrict__ Bg,
    const float* __restrict__ wsE, float* __restrict__ wsResNorm,
    int Pg, int goff, int koff)
{
    __shared__ float sNorm[BATCH];
    const int tid = threadIdx.x;
    if (tid < BATCH) sNorm[tid] = 0.f;
    __syncthreads();

    const int w    = tid >> 5;
    const int lane = tid & 31;
    const int mt   = w & 3;
    const int half = w >> 2;
    const int lo   = lane & 15;
    const int hi   = lane >> 4;
    const int pbase = blockIdx.x * 64;

    v8f acc[2] = { v8f_zero(), v8f_zero() };
    backproj_acc<KG>(wsE, Bg, Pg, koff, pbase, mt, half, lo, hi, acc);

#pragma unroll
    for (int t = 0; t < 2; ++t) {
        int nt = half + 2 * t;
        int p  = pbase + nt * 16 + lo;
        bool valid = (p < Pg);
        unsigned gcol = (unsigned)(goff + p);
#pragma unroll
        for (int v = 0; v < 8; ++v) {
            int row = mt * 16 + v + hi * 8;
            float g = valid ? G[(unsigned)row * PTOT + gcol] : 0.f;
            float r = valid ? (g - acc[t][v]) : 0.f;   // acc garbage if !valid
            float rr = r * r;
            rr += __shfl_xor(rr, 1, 16);
            rr += __shfl_xor(rr, 2, 16);
            rr += __shfl_xor(rr, 4, 16);
            rr += __shfl_xor(rr, 8, 16);
            if (lo == 0) atomicAdd(&sNorm[row], rr);  // lanes 0 and 16
        }
    }
    __syncthreads();
    if (tid < BATCH) atomicAdd(&wsResNorm[tid], sNorm[tid]);
}

// ---------------------------------------------------------------------------
// Phase 4: per-row clip scales
// ---------------------------------------------------------------------------
__global__ void k_scale(const float* __restrict__ wsResNorm,
                        float* __restrict__ wsScale)
{
    int t = threadIdx.x;
    if (t < BATCH) wsScale[t] = fminf(CLIP1 / sqrtf(wsResNorm[t]), 1.0f);
}

// ---------------------------------------------------------------------------
// Phase 5: avg clipped residual + avg target grad (pass 2 of back-projection)
// ---------------------------------------------------------------------------
template <int KG>
__global__ __launch_bounds__(256) void k_resout(
    const float* __restrict__ G, const float* __restrict__ Bg,
    const float* __restrict__ wsE, const float* __restrict__ wsScale,
    float* __restrict__ out,
    int Pg, int goff, int koff)
{
    __shared__ float sRes[64];
    __shared__ float sTg[64];
    __shared__ float sScale[BATCH];
    const int tid = threadIdx.x;
    if (tid < 64) { sRes[tid] = 0.f; sTg[tid] = 0.f; sScale[tid] = wsScale[tid]; }
    __syncthreads();

    const int w    = tid >> 5;
    const int lane = tid & 31;
    const int mt   = w & 3;
    const int half = w >> 2;
    const int lo   = lane & 15;
    const int hi   = lane >> 4;
    const int pbase = blockIdx.x * 64;

    v8f acc[2] = { v8f_zero(), v8f_zero() };
    backproj_acc<KG>(wsE, Bg, Pg, koff, pbase, mt, half, lo, hi, acc);

#pragma unroll
    for (int t = 0; t < 2; ++t) {
        int nt = half + 2 * t;
        int p  = pbase + nt * 16 + lo;
        bool valid = (p < Pg);
        unsigned gcol = (unsigned)(goff + p);
        float cres = 0.f, ctg = 0.f;
#pragma unroll
        for (int v = 0; v < 8; ++v) {
            int row = mt * 16 + v + hi * 8;
            float g = valid ? G[(unsigned)row * PTOT + gcol] : 0.f;
            if (valid) cres += sScale[row] * (g - acc[t][v]);
            ctg += g;
        }
        // combine the two half-wave row groups (same column)
        cres += __shfl_xor(cres, 16, 32);
        ctg  += __shfl_xor(ctg, 16, 32);
        if (hi == 0) {
            atomicAdd(&sRes[nt * 16 + lo], cres);
            atomicAdd(&sTg[nt * 16 + lo], ctg);
        }
    }
    __syncthreads();
    if (tid < 64) {
        int p = pbase + tid;
        if (p < Pg) {
            out[KTOT + goff + p]        = sRes[tid] * (1.0f / BATCH);
            out[KTOT + PTOT + goff + p] = sTg[tid]  * (1.0f / BATCH);
        }
    }
}

// ---------------------------------------------------------------------------
extern "C" void kernel_launch(void* const* d_in, const int* in_sizes, int n_in,
                              void* d_out, int out_size, void* d_ws, size_t ws_size,
                              hipStream_t stream)
{
    const float* G  = (const float*)d_in[0];
    const float* B0 = (const float*)d_in[1];
    const float* B1 = (const float*)d_in[2];
    const float* B2 = (const float*)d_in[3];
    float* out = (float*)d_out;
    float* ws  = (float*)d_ws;
    float* wsE       = ws + WS_E;
    float* wsResNorm = ws + WS_RESNORM;
    float* wsScale   = ws + WS_RESSCALE;

    k_init<<<(WS_FLOATS + 255) / 256, 256, 0, stream>>>(ws);

    const int CHUNK = 2048;
    k_gemm1<104><<<(100000 + CHUNK - 1) / CHUNK, 256, 0, stream>>>(
        G, B0, wsE, 100000, 0, 0, CHUNK);
    k_gemm1<156><<<(225000 + CHUNK - 1) / CHUNK, 256, 0, stream>>>(
        G, B1, wsE, 225000, 100000, 104, CHUNK);
    k_gemm1<138><<<(175000 + CHUNK - 1) / CHUNK, 256, 0, stream>>>(
        G, B2, wsE, 175000, 325000, 260, CHUNK);

    k_embfinish<<<1, 256, 0, stream>>>(wsE, out);

    k_resnorm<104><<<(100000 + 63) / 64, 256, 0, stream>>>(
        G, B0, wsE, wsResNorm, 100000, 0, 0);
    k_resnorm<156><<<(225000 + 63) / 64, 256, 0, stream>>>(
        G, B1, wsE, wsResNorm, 225000, 100000, 104);
    k_resnorm<138><<<(175000 + 63) / 64, 256, 0, stream>>>(
        G, B2, wsE, wsResNorm, 175000, 325000, 260);

    k_scale<<<1, 64, 0, stream>>>(wsResNorm, wsScale);

    k_resout<104><<<(100000 + 63) / 64, 256, 0, stream>>>(
        G, B0, wsE, wsScale, out, 100000, 0, 0);
    k_resout<156><<<(225000 + 63) / 64, 256, 0, stream>>>(
        G, B1, wsE, wsScale, out, 225000, 100000, 104);
    k_resout<138><<<(175000 + 63) / 64, 256, 0, stream>>>(
        G, B2, wsE, wsScale, out, 175000, 325000, 260);
}